// RBFController_10866267258984
// MI455X (gfx1250) — compile-verified
//
#include <hip/hip_runtime.h>
#include <math.h>

// RBF controller forward for MI455X (gfx1250, wave32).
// One wave (32 lanes) per batch item. Dense 16x16-shaped matmuls go through
// V_WMMA_F32_16X16X4_F32 (f32 tensor path). 16x16 SPD inverses via LDS
// Gauss-Jordan. Pairwise-center quadratic forms are algebraically expanded
// (centers constant across batch) so the 1024-RHS solve of the reference
// collapses to one 32x32 Gram matrix per batch item.
//
// WMMA f32 16x16x4 VGPR layout assumptions (ISA 7.12.2):
//   A (16x4):  lane = 16*half + M, reg v holds element (M, 2*half + v)
//   B (4x16):  lane = 16*half + N, reg v holds element (2*half + v, N)
//   C/D(16x16):lane = 16*half + N, reg v holds element (v + 8*half, N)

#define BATCH 8192
#define DDIM  16
#define NCTR  32
#define KACT  8

typedef float v2f __attribute__((ext_vector_type(2)));
typedef float v8f __attribute__((ext_vector_type(8)));

__device__ __forceinline__ v2f fragA(const float* P, int rs, int cs) {
  int ln = threadIdx.x & 31; int lo = ln & 15; int kb = (ln >> 4) * 2;
  v2f a; a.x = P[lo * rs + kb * cs]; a.y = P[lo * rs + (kb + 1) * cs]; return a;
}
__device__ __forceinline__ v2f fragB(const float* P, int rs, int cs) {
  int ln = threadIdx.x & 31; int lo = ln & 15; int kb = (ln >> 4) * 2;
  v2f b; b.x = P[kb * rs + lo * cs]; b.y = P[(kb + 1) * rs + lo * cs]; return b;
}
__device__ __forceinline__ void storeD(float* P, int rs, v8f d) {
  int ln = threadIdx.x & 31; int lo = ln & 15; int mh = (ln >> 4) * 8;
#pragma unroll
  for (int v = 0; v < 8; ++v) P[(v + mh) * rs + lo] = d[v];
}

// D(16x16) = A(16xK) * B(Kx16); ars/acs = row/col strides of A (acs = K stride),
// brs/bcs = strides of B (brs = K stride). Transposed operands via swapped strides.
template <int K>
__device__ __forceinline__ v8f mm_tile(const float* A, int ars, int acs,
                                       const float* B, int brs, int bcs) {
  v8f c = {};
#pragma unroll
  for (int k0 = 0; k0 < K; k0 += 4) {
    v2f a = fragA(A + k0 * acs, ars, acs);
    v2f b = fragB(B + k0 * brs, brs, bcs);
    c = __builtin_amdgcn_wmma_f32_16x16x4_f32(false, a, false, b, (short)0, c,
                                              false, false);
  }
  return c;
}

// In-place Gauss-Jordan on [A | I] (16x32 in aug, lane owns column ln).
// SPD input -> no pivoting. Returns det (pivot product) and sum of log pivots.
__device__ void gauss_jordan16(float (*aug)[33], float& det, float& logdet) {
  const int ln = threadIdx.x;
  det = 1.f; logdet = 0.f;
  for (int k = 0; k < 16; ++k) {
    float piv = aug[k][k];
    det *= piv;
    logdet += __logf(piv);
    float ip = 1.f / piv;
    float bk = aug[k][ln] * ip;
    aug[k][ln] = bk;
    __syncthreads();
#pragma unroll
    for (int i = 0; i < 16; ++i) {
      if (i == k) continue;
      float f = aug[i][k];       // wave-uniform read precedes per-lane store
      aug[i][ln] -= f * bk;
    }
    __syncthreads();
  }
}

__global__ __launch_bounds__(32) void rbf_ctrl_kernel(
    const float* __restrict__ gM, const float* __restrict__ gS,
    const float* __restrict__ gC, const float* __restrict__ gW,
    const float* __restrict__ gL, float* __restrict__ outM,
    float* __restrict__ outS, float* __restrict__ outV) {
  const int b = blockIdx.x;
  const int ln = threadIdx.x;

  __shared__ float sS[16][17];     // input covariance
  __shared__ float sAug[16][33];   // Gauss-Jordan workspace
  __shared__ float sA1p[16][17];   // D * Bm^{-1} * D
  __shared__ float sA[16][17];     // Bq^{-1}
  __shared__ float sC[32][17];     // centers
  __shared__ float sWp[32][17];    // weights padded to 16 cols
  __shared__ float sQ[32][33];
  __shared__ float sT1[32][17];    // C @ A1p
  __shared__ float sT[32][17];     // C @ A
  __shared__ float sG[32][33];     // C A C^T
  __shared__ float sR[32][17];     // Q @ Wp
  __shared__ float sP[16][17];
  __shared__ float sVm[16][17];
  __shared__ float sS16[16][17];
  __shared__ float sm[16], sInvL[16], sInvLam[16], sLen[16];
  __shared__ float svec[16];
  __shared__ float sr1[32], sr[32], sq1[32], sphi[32];
  __shared__ float sMpre[8], sDg[8], sMo[8], sCd[8];

  // ---- loads (coalesced) ----
  if (ln < 16) {
    float l = gL[ln];
    sLen[ln] = l; sInvL[ln] = 1.f / l; sInvLam[ln] = 1.f / (l * l);
    sm[ln] = gM[b * 16 + ln];
  }
#pragma unroll
  for (int t = 0; t < 16; ++t) { int i = t * 32 + ln; sC[i >> 4][i & 15] = gC[i]; }
#pragma unroll
  for (int t = 0; t < 8; ++t) { int i = t * 32 + ln; sWp[i >> 3][i & 7] = gW[i]; }
#pragma unroll
  for (int k = 8; k < 16; ++k) sWp[ln][k] = 0.f;
#pragma unroll
  for (int t = 0; t < 8; ++t) {
    int i = t * 32 + ln;
    sS[i >> 4][i & 15] = gS[b * 256 + i];
  }
  __syncthreads();

  // ---- Bm = iL S iL + I  ->  A1 = Bm^{-1}, detBm ----
#pragma unroll
  for (int i = 0; i < 16; ++i) {
    float v;
    if (ln < 16) v = sS[i][ln] * sInvL[i] * sInvL[ln] + (i == ln ? 1.f : 0.f);
    else         v = ((ln - 16) == i) ? 1.f : 0.f;
    sAug[i][ln] = v;
  }
  __syncthreads();
  float detBm, ldBm;
  gauss_jordan16(sAug, detBm, ldBm);
  {
    int lo = ln & 15, hf = ln >> 4;
#pragma unroll
    for (int t = 0; t < 8; ++t) {
      int i = hf * 8 + t;
      sA1p[i][lo] = sAug[i][16 + lo] * sInvL[i] * sInvL[lo];
    }
  }
  __syncthreads();

  // ---- Bq = S + diag(l^2/2) -> A = Bq^{-1}, logdetBq ----
#pragma unroll
  for (int i = 0; i < 16; ++i) {
    float v;
    if (ln < 16) v = sS[i][ln] + (i == ln ? 0.5f * sLen[i] * sLen[i] : 0.f);
    else         v = ((ln - 16) == i) ? 1.f : 0.f;
    sAug[i][ln] = v;
  }
  __syncthreads();
  float detBq, ldBq;
  gauss_jordan16(sAug, detBq, ldBq);
  {
    int lo = ln & 15, hf = ln >> 4;
#pragma unroll
    for (int t = 0; t < 8; ++t) { int i = hf * 8 + t; sA[i][lo] = sAug[i][16 + lo]; }
  }
  __syncthreads();

  // ---- WMMA: T1 = C@A1p, T = C@A (two row tiles each) ----
  {
    v8f d0 = mm_tile<16>(&sC[0][0], 17, 1, &sA1p[0][0], 17, 1);
    v8f d1 = mm_tile<16>(&sC[16][0], 17, 1, &sA1p[0][0], 17, 1);
    storeD(&sT1[0][0], 17, d0);
    storeD(&sT1[16][0], 17, d1);
    v8f e0 = mm_tile<16>(&sC[0][0], 17, 1, &sA[0][0], 17, 1);
    v8f e1 = mm_tile<16>(&sC[16][0], 17, 1, &sA[0][0], 17, 1);
    storeD(&sT[0][0], 17, e0);
    storeD(&sT[16][0], 17, e1);
  }
  __syncthreads();
  // ---- WMMA: G = T @ C^T (4 tiles), C^T read with swapped strides ----
  {
    v8f g00 = mm_tile<16>(&sT[0][0], 17, 1, &sC[0][0], 1, 17);
    v8f g01 = mm_tile<16>(&sT[0][0], 17, 1, &sC[16][0], 1, 17);
    v8f g10 = mm_tile<16>(&sT[16][0], 17, 1, &sC[0][0], 1, 17);
    v8f g11 = mm_tile<16>(&sT[16][0], 17, 1, &sC[16][0], 1, 17);
    storeD(&sG[0][0], 33, g00);
    storeD(&sG[0][16], 33, g01);
    storeD(&sG[16][0], 33, g10);
    storeD(&sG[16][16], 33, g11);
  }
  __syncthreads();

  // ---- quadratic-form vectors: v1 = A1p m ; r1, q1 ; v = A m ; r ----
  if (ln < 16) {
    float a = 0.f;
#pragma unroll
    for (int j = 0; j < 16; ++j) a += sA1p[ln][j] * sm[j];
    svec[ln] = a;
  }
  __syncthreads();
  float mA1m = 0.f;
#pragma unroll
  for (int d = 0; d < 16; ++d) mA1m += sm[d] * svec[d];
  {
    float a = 0.f, qq = 0.f;
#pragma unroll
    for (int d = 0; d < 16; ++d) {
      a += sC[ln][d] * svec[d];
      qq += sT1[ln][d] * sC[ln][d];
    }
    sr1[ln] = a; sq1[ln] = qq;
  }
  __syncthreads();
  if (ln < 16) {
    float a = 0.f;
#pragma unroll
    for (int j = 0; j < 16; ++j) a += sA[ln][j] * sm[j];
    svec[ln] = a;
  }
  __syncthreads();
  float mAm = 0.f;
#pragma unroll
  for (int d = 0; d < 16; ++d) mAm += sm[d] * svec[d];
  {
    float a = 0.f;
#pragma unroll
    for (int d = 0; d < 16; ++d) a += sC[ln][d] * svec[d];
    sr[ln] = a;
  }
  __syncthreads();

  // ---- phi, pre-squash mean ----
  float c_coef = 1.0f /*VARIANCE*/ / sqrtf(detBm);
  sphi[ln] = c_coef * __expf(-0.5f * (sq1[ln] - 2.f * sr1[ln] + mA1m));
  __syncthreads();
  if (ln < 8) {
    float a = 0.f;
#pragma unroll
    for (int n = 0; n < 32; ++n) a += sphi[n] * sWp[n][ln];
    sMpre[ln] = a;
  }
  __syncthreads();

  // ---- Q matrix (lane = column) ----
  float ldLh = 0.f;
#pragma unroll
  for (int d = 0; d < 16; ++d) ldLh += __logf(0.5f * sLen[d] * sLen[d]);
  float qscale = 1.0f /*VARIANCE^2*/ * __expf(0.5f * (ldLh - ldBq));
  {
    const int n2 = ln;
    float gd2 = sG[n2][n2], rr2 = sr[n2];
#pragma unroll 4
    for (int n = 0; n < 32; ++n) {
      float s1 = 0.f;
#pragma unroll
      for (int d = 0; d < 16; ++d) {
        float df = sC[n][d] - sC[n2][d];
        s1 += df * df * sInvLam[d];
      }
      float e1 = -0.25f * s1;
      float e2 = -0.5f * (0.25f * (sG[n][n] + 2.f * sG[n][n2] + gd2) -
                          (sr[n] + rr2) + mAm);
      sQ[n][n2] = qscale * __expf(e1 + e2);
    }
  }
  __syncthreads();

  // ---- WMMA: R = Q@Wp (K=32, two row tiles), S16 = Wp^T @ R (K=32) ----
  {
    v8f r0 = mm_tile<32>(&sQ[0][0], 33, 1, &sWp[0][0], 17, 1);
    v8f r1_ = mm_tile<32>(&sQ[16][0], 33, 1, &sWp[0][0], 17, 1);
    storeD(&sR[0][0], 17, r0);
    storeD(&sR[16][0], 17, r1_);
  }
  __syncthreads();
  {
    v8f s0 = mm_tile<32>(&sWp[0][0], 1, 17, &sR[0][0], 17, 1);  // Wp^T
    storeD(&sS16[0][0], 17, s0);
  }

  // ---- P = C^T (phi .* W) - m Mpre^T (cols 8..15 zero) ----
#pragma unroll
  for (int t = 0; t < 8; ++t) {
    int idx = ln * 8 + t;
    int d = idx >> 4, j = idx & 15;
    float v = 0.f;
    if (j < 8) {
      for (int n = 0; n < 32; ++n) v += sC[n][d] * sphi[n] * sWp[n][j];
      v -= sm[d] * sMpre[j];
    }
    sP[d][j] = v;
  }
  __syncthreads();
  // ---- WMMA: V = A1p @ P ----
  {
    v8f vt = mm_tile<16>(&sA1p[0][0], 17, 1, &sP[0][0], 17, 1);
    storeD(&sVm[0][0], 17, vt);
  }
  __syncthreads();

  // ---- symmetrize S, add jitter ----
  {
    int i0 = ln * 2, i1 = ln * 2 + 1;
    int k0 = i0 >> 3, l0 = i0 & 7, k1 = i1 >> 3, l1 = i1 & 7;
    float v0 = 0.5f * (sS16[k0][l0] + sS16[l0][k0]) - sMpre[k0] * sMpre[l0] +
               (k0 == l0 ? 1e-6f : 0.f);
    float v1 = 0.5f * (sS16[k1][l1] + sS16[l1][k1]) - sMpre[k1] * sMpre[l1] +
               (k1 == l1 ? 1e-6f : 0.f);
    __syncthreads();
    sS16[k0][l0] = v0;
    sS16[k1][l1] = v1;
  }
  __syncthreads();

  // ---- squash (MAX_ACTION = 1) ----
  if (ln < 8) {
    float dg = sS16[ln][ln];
    sDg[ln] = dg;
    float ex = __expf(-0.5f * dg);
    sMo[ln] = ex * __sinf(sMpre[ln]);
    sCd[ln] = ex * __cosf(sMpre[ln]);
  }
  __syncthreads();

  // ---- outputs ----
  if (ln < 8) outM[b * 8 + ln] = sMo[ln];
#pragma unroll
  for (int t = 0; t < 2; ++t) {
    int idx = ln * 2 + t;
    int k = idx >> 3, l = idx & 7;
    float skl = sS16[k][l];
    float lq = -0.5f * (sDg[k] + sDg[l]);
    float q = __expf(lq);
    float md = sMpre[k] - sMpre[l], ms = sMpre[k] + sMpre[l];
    float val = 0.5f * ((__expf(lq + skl) - q) * __cosf(md) -
                        (__expf(lq - skl) - q) * __cosf(ms));
    outS[b * 64 + idx] = val;
  }
#pragma unroll
  for (int t = 0; t < 4; ++t) {
    int idx = ln * 4 + t;
    int d = idx >> 3, l = idx & 7;
    outV[b * 128 + idx] = sVm[d][l] * sCd[l];
  }
}

extern "C" void kernel_launch(void* const* d_in, const int* in_sizes, int n_in,
                              void* d_out, int out_size, void* d_ws,
                              size_t ws_size, hipStream_t stream) {
  (void)in_sizes; (void)n_in; (void)d_ws; (void)ws_size; (void)out_size;
  const float* gM = (const float*)d_in[0];
  const float* gS = (const float*)d_in[1];
  const float* gC = (const float*)d_in[2];
  const float* gW = (const float*)d_in[3];
  const float* gL = (const float*)d_in[4];
  float* outM = (float*)d_out;                       // (B, K)
  float* outS = outM + (size_t)BATCH * KACT;         // (B, K, K)
  float* outV = outS + (size_t)BATCH * KACT * KACT;  // (B, D, K)
  rbf_ctrl_kernel<<<BATCH, 32, 0, stream>>>(gM, gS, gC, gW, gL, outM, outS,
                                            outV);
}